// QuanvolutionHybridGraphQL_65481071405504
// MI455X (gfx1250) — compile-verified
//
#include <hip/hip_runtime.h>
#include <math.h>

// ---------------------------------------------------------------------------
// QuanvolutionHybridGraphQL for MI455X (gfx1250, wave32)
//
// Outputs (concatenated in d_out, fp32):
//   out : [4096, 10]            (offset 0)
//   adj : [4096, 196, 196]      (offset 40960)
//
// adj is 630 MB of writes -> store-BW bound (~27 us floor @ 23.3 TB/s).
// Gram matrix inner dim K=4 == V_WMMA_F32_16X16X4_F32's exact shape.
// Interior 12x12 tile grid is branch-free; the 4-row/4-col edge strip is
// handled with plain VALU dots so no store needs exec-mask predication.
// ---------------------------------------------------------------------------

typedef float v2f __attribute__((ext_vector_type(2)));
typedef float v8f __attribute__((ext_vector_type(8)));

#define NPATCH 196
#define NTILES 13                 // ceil(196/16)
#define NINNER 12                 // full tiles per dimension (rows 0..191)
#define PADP   (NTILES * 16)      // 208 (pad rows are zero vectors)
#define NHID   32
#define NCLS   10
#define BATCH  4096
#define THRESH 0.8f

// RY(v+0.5) circuit with CX chain -> <Z_k> = prod_{i<=k} cos(v_i + 0.5)
__device__ __forceinline__ float4 qvec(const float4 p) {
  const float e0 = cosf(p.x + 0.5f);
  const float e1 = e0 * cosf(p.y + 0.5f);
  const float e2 = e1 * cosf(p.z + 0.5f);
  const float e3 = e2 * cosf(p.w + 0.5f);
  return make_float4(e0, e1, e2, e3);
}

// One block per image: build normalized vectors in LDS, then Gram tiles.
__global__ __launch_bounds__(256) void quanv_adj_kernel(
    const float* __restrict__ x,    // [B, 784] (== [B,196,4] patches)
    float* __restrict__ adj)        // [B, 196, 196]
{
  __shared__ float an[PADP][4];     // normalized vectors, zero-padded

  const int b   = blockIdx.x;
  const int tid = threadIdx.x;

  // ---- phase 1: per-patch quantum vectors, normalized, into LDS ----
  if (tid < PADP) {
    float4 v = make_float4(0.f, 0.f, 0.f, 0.f);
    if (tid < NPATCH) {
      const float4 p = *(const float4*)(x + (size_t)b * (NPATCH * 4) + tid * 4);
      const float4 e = qvec(p);
      const float inv =
          1.0f / (sqrtf(e.x * e.x + e.y * e.y + e.z * e.z + e.w * e.w) + 1e-12f);
      v = make_float4(e.x * inv, e.y * inv, e.z * inv, e.w * inv);
    }
    an[tid][0] = v.x; an[tid][1] = v.y; an[tid][2] = v.z; an[tid][3] = v.w;
  }
  __syncthreads();

  const int lane = tid & 31;
  const int wave = tid >> 5;        // 8 waves per block
  const int half = lane >> 4;       // 0 -> K={0,1}, 1 -> K={2,3}
  const int l16  = lane & 15;

  float* __restrict__ adjb = adj + (size_t)b * NPATCH * NPATCH;

  // ---- phase 2a: interior 12x12 tiles via WMMA (144 tiles, 18 per wave) ----
  // Wave-uniform loop; EXEC stays all-ones at every WMMA (ISA requirement),
  // and every store is in-range -> fully branch-free hot path.
  for (int t = wave; t < NINNER * NINNER; t += 8) {
    const int ti = t / NINNER;
    const int tj = t % NINNER;

    // A (16x4) and B (4x16 = an^T) share the same per-lane layout:
    // lane L<16 holds row (base+L) K=0,1; lane L>=16 holds row (base+L-16)
    // K=2,3.  One ds_load_b64 per fragment.
    const v2f afrag = *(const v2f*)&an[ti * 16 + l16][2 * half];
    const v2f bfrag = *(const v2f*)&an[tj * 16 + l16][2 * half];

    v8f c = {};
    c = __builtin_amdgcn_wmma_f32_16x16x4_f32(
        /*neg_a=*/false, afrag, /*neg_b=*/false, bfrag,
        /*c_mod=*/(short)0, c, /*reuse_a=*/false, /*reuse_b=*/false);

    // C/D layout: VGPR m -> M = m + 8*half, N = l16.
    const int col  = tj * 16 + l16;
    const int row0 = ti * 16 + 8 * half;
    #pragma unroll
    for (int m = 0; m < 8; ++m) {
      const int row = row0 + m;
      const float d = c[m];
      // branchless diagonal mask + threshold (cndmask, no exec churn)
      const float v = (d * d >= THRESH && row != col) ? 1.0f : 0.0f;
      // adj is 630MB (>> 192MB L2): stream with non-temporal stores.
      __builtin_nontemporal_store(v, &adjb[row * NPATCH + col]);
    }
  }

  // ---- phase 2b: edge strip (row>=192 or col>=192) via plain VALU dots ----
  // Region A: rows 192..195, cols 0..195 (784 elems, coalesced by col)
  for (int idx = tid; idx < 4 * NPATCH; idx += 256) {
    const int row = 192 + idx / NPATCH;
    const int col = idx % NPATCH;
    const float d = an[row][0] * an[col][0] + an[row][1] * an[col][1] +
                    an[row][2] * an[col][2] + an[row][3] * an[col][3];
    const float v = (d * d >= THRESH && row != col) ? 1.0f : 0.0f;
    __builtin_nontemporal_store(v, &adjb[row * NPATCH + col]);
  }
  // Region B: rows 0..191, cols 192..195 (768 elems, 3 per thread exactly)
  for (int idx = tid; idx < NINNER * 16 * 4; idx += 256) {
    const int row = idx >> 2;
    const int col = 192 + (idx & 3);
    const float d = an[row][0] * an[col][0] + an[row][1] * an[col][1] +
                    an[row][2] * an[col][2] + an[row][3] * an[col][3];
    const float v = (d * d >= THRESH) ? 1.0f : 0.0f;   // row<192 => row!=col
    __builtin_nontemporal_store(v, &adjb[row * NPATCH + col]);
  }
}

// One wave (32 lanes) per image: pooled mean -> 32-wide MLP -> 10 logits.
__global__ __launch_bounds__(256) void head_kernel(
    const float* __restrict__ x,    // [B, 784]
    const float* __restrict__ W1,   // [32, 4]
    const float* __restrict__ b1,   // [32]
    const float* __restrict__ W2,   // [10, 32]
    const float* __restrict__ b2,   // [10]
    float* __restrict__ out)        // [B, 10]
{
  const int gtid = blockIdx.x * blockDim.x + threadIdx.x;
  const int b    = gtid >> 5;       // one wave32 per image
  const int lane = threadIdx.x & 31;
  if (b >= BATCH) return;

  const float* xb = x + (size_t)b * (NPATCH * 4);

  // strided accumulation of UN-normalized cumprod vectors (pooled mean)
  float s0 = 0.f, s1 = 0.f, s2 = 0.f, s3 = 0.f;
  for (int p = lane; p < NPATCH; p += 32) {
    const float4 e = qvec(*(const float4*)(xb + p * 4));
    s0 += e.x; s1 += e.y; s2 += e.z; s3 += e.w;
  }
  // butterfly all-reduce across the wave32
  #pragma unroll
  for (int m = 16; m >= 1; m >>= 1) {
    s0 += __shfl_xor(s0, m, 32);
    s1 += __shfl_xor(s1, m, 32);
    s2 += __shfl_xor(s2, m, 32);
    s3 += __shfl_xor(s3, m, 32);
  }
  const float invN = 1.0f / (float)NPATCH;
  const float p0 = s0 * invN, p1 = s1 * invN, p2 = s2 * invN, p3 = s3 * invN;

  if (lane < NCLS) {
    float o = b2[lane];
    #pragma unroll
    for (int h = 0; h < NHID; ++h) {
      const float hv = b1[h] + p0 * W1[h * 4 + 0] + p1 * W1[h * 4 + 1] +
                               p2 * W1[h * 4 + 2] + p3 * W1[h * 4 + 3];
      o += hv * W2[lane * NHID + h];
    }
    out[(size_t)b * NCLS + lane] = o;
  }
}

extern "C" void kernel_launch(void* const* d_in, const int* in_sizes, int n_in,
                              void* d_out, int out_size, void* d_ws, size_t ws_size,
                              hipStream_t stream) {
  (void)in_sizes; (void)n_in; (void)out_size; (void)d_ws; (void)ws_size;

  const float* x  = (const float*)d_in[0];   // [4096,1,28,28]
  const float* W1 = (const float*)d_in[1];   // [32,4]
  const float* b1 = (const float*)d_in[2];   // [32]
  const float* W2 = (const float*)d_in[3];   // [10,32]
  const float* b2 = (const float*)d_in[4];   // [10]

  float* out = (float*)d_out;                       // [4096,10]
  float* adj = out + (size_t)BATCH * NCLS;          // [4096,196,196]

  // 512 blocks * 8 waves = 4096 images
  head_kernel<<<BATCH / 8, 256, 0, stream>>>(x, W1, b1, W2, b2, out);
  // one block per image for the adjacency
  quanv_adj_kernel<<<BATCH, 256, 0, stream>>>(x, adj);
}